// StructuralGNN_36885179138565
// MI455X (gfx1250) — compile-verified
//
#include <hip/hip_runtime.h>
#include <math.h>

#define N_NODES 50000
#define N_EDGES 600000
#define HD 128
#define NFEAT 12
#define NL 4
#define TN 32      // nodes per block (2 waves x 16-row tiles)
#define TLD 132    // padded LDS row stride for 128 cols
#define T2LD 68    // padded LDS row stride for 64 cols

typedef __attribute__((ext_vector_type(2))) float v2f;
typedef __attribute__((ext_vector_type(8))) float v8f;

// acc[NJ] += A(16x128, LDS, row-major stride lda) @ W(128 x 16*NJ, global row-major stride ldw)
// f32 WMMA operand layout (16x16x4):
//   A: lane m=lane&15 holds row m; VGPR0/1 = K = 2*kh, 2*kh+1 (kh = lane>>4)
//   B: lane n=lane&15 holds col n; VGPR0/1 = K rows 2*kh, 2*kh+1
template <int NJ>
__device__ __forceinline__ void wmma_gemm_f32(const float* Alds, int lda,
                                              const float* __restrict__ W, int ldw,
                                              v8f* acc) {
  const int lane = threadIdx.x & 31;
  const int m = lane & 15;
  const int kh = lane >> 4;
  const float* arow = Alds + m * lda + 2 * kh;
  const float* wcol = W + 2 * kh * ldw + m;
#pragma unroll 2
  for (int k = 0; k < HD; k += 4) {
    v2f a;
    a.x = arow[k];
    a.y = arow[k + 1];
    const float* wp = wcol + k * ldw;
#pragma unroll
    for (int j = 0; j < NJ; ++j) {
      v2f b;
      b.x = wp[j * 16];
      b.y = wp[j * 16 + ldw];
      acc[j] = __builtin_amdgcn_wmma_f32_16x16x4_f32(false, a, false, b,
                                                     (short)0, acc[j], false, false);
    }
  }
}

// ---------------- ratio = max ||x[:,3:6]|| ----------------
__global__ void ratio_kernel(const float* __restrict__ x, float* __restrict__ ratio_ws) {
  int n = blockIdx.x * blockDim.x + threadIdx.x;
  float v = 0.f;
  if (n < N_NODES) {
    float a = x[n * NFEAT + 3], b = x[n * NFEAT + 4], c = x[n * NFEAT + 5];
    v = sqrtf(a * a + b * b + c * c);
  }
#pragma unroll
  for (int off = 16; off; off >>= 1) v = fmaxf(v, __shfl_xor(v, off, 32));
  if ((threadIdx.x & 31) == 0)
    atomicMax((unsigned int*)ratio_ws, __float_as_uint(v));  // nonneg floats: bit order == value order
}

// ---------------- in-degree ----------------
__global__ void degree_kernel(const int* __restrict__ ei, float* __restrict__ deg) {
  int e = blockIdx.x * blockDim.x + threadIdx.x;
  if (e < N_EDGES) atomicAdd(&deg[ei[N_EDGES + e]], 1.0f);
}

// ---------------- edge scatter: one wave per edge ----------------
__global__ void scatter_kernel(const float* __restrict__ h, const int* __restrict__ ei,
                               float* __restrict__ agg) {
  int w = (blockIdx.x * blockDim.x + threadIdx.x) >> 5;
  int lane = threadIdx.x & 31;
  if (w >= N_EDGES) return;
  int s = ei[w];
  int d = ei[N_EDGES + w];
  const float4 v = *(const float4*)(h + (size_t)s * HD + lane * 4);
  float* a = agg + (size_t)d * HD + lane * 4;
  atomicAdd(a + 0, v.x);
  atomicAdd(a + 1, v.y);
  atomicAdd(a + 2, v.z);
  atomicAdd(a + 3, v.w);
}

// ---------------- encoder: h = relu(x@W1+b1)@W2+b2 ----------------
__global__ __launch_bounds__(64) void encoder_kernel(
    const float* __restrict__ x, const float* __restrict__ W1, const float* __restrict__ b1,
    const float* __restrict__ W2, const float* __restrict__ b2, float* __restrict__ h) {
  __shared__ float W1s[NFEAT * HD];
  __shared__ float b1s[HD];
  __shared__ float Xs[TN * NFEAT];
  __shared__ float Ts[TN * TLD];
  const int tid = threadIdx.x;
  const int base = blockIdx.x * TN;
  const bool full = (base + TN) <= N_NODES;
  for (int i = tid; i < NFEAT * HD; i += 64) W1s[i] = W1[i];
  for (int i = tid; i < HD; i += 64) b1s[i] = b1[i];
  if (full) {
    for (int i = tid; i < TN * NFEAT; i += 64) Xs[i] = x[base * NFEAT + i];
  } else {
    for (int i = tid; i < TN * NFEAT; i += 64) {
      int gi = base * NFEAT + i;
      Xs[i] = (gi < N_NODES * NFEAT) ? x[gi] : 0.f;
    }
  }
  __syncthreads();
  for (int idx = tid; idx < TN * HD; idx += 64) {
    int r = idx >> 7, c = idx & 127;
    float acc = b1s[c];
#pragma unroll
    for (int f = 0; f < NFEAT; ++f) acc = fmaf(Xs[r * NFEAT + f], W1s[f * HD + c], acc);
    Ts[r * TLD + c] = fmaxf(acc, 0.f);
  }
  __syncthreads();
  const int lane = tid & 31, wv = tid >> 5;
  const int m = lane & 15, kh = lane >> 4;
  v8f acc[8];
#pragma unroll
  for (int j = 0; j < 8; ++j) {
    float bv = b2[j * 16 + m];
#pragma unroll
    for (int i = 0; i < 8; ++i) acc[j][i] = bv;
  }
  wmma_gemm_f32<8>(Ts + wv * 16 * TLD, TLD, W2, HD, acc);
  if (full) {
#pragma unroll
    for (int i = 0; i < 8; ++i) {
      size_t node = base + wv * 16 + i + 8 * kh;
#pragma unroll
      for (int j = 0; j < 8; ++j) h[node * HD + j * 16 + m] = acc[j][i];
    }
  } else {
#pragma unroll
    for (int i = 0; i < 8; ++i) {
      int node = base + wv * 16 + i + 8 * kh;
      if (node < N_NODES)
#pragma unroll
        for (int j = 0; j < 8; ++j) h[(size_t)node * HD + j * 16 + m] = acc[j][i];
    }
  }
}

// ---------------- SAGE layer: y = (agg/deg)@Wl + bl + h@Wr ; LN ; h += relu(y) ----------------
__global__ __launch_bounds__(64) void sage_kernel(
    const float* __restrict__ hin, const float* __restrict__ agg, const float* __restrict__ deg,
    const float* __restrict__ Wl, const float* __restrict__ bl, const float* __restrict__ Wr,
    const float* __restrict__ lng, const float* __restrict__ lnb, float* __restrict__ hout) {
  __shared__ float As[TN * TLD];
  __shared__ float Hs[TN * TLD];
  __shared__ float invd[TN];
  const int tid = threadIdx.x;
  const int base = blockIdx.x * TN;
  const bool full = (base + TN) <= N_NODES;
  if (tid < TN) {
    int node = base + tid;
    float d = (full || node < N_NODES) ? deg[node] : 1.f;
    invd[tid] = 1.f / fmaxf(d, 1.f);
  }
  if (full) {
    for (int idx = tid; idx < TN * HD; idx += 64) {
      int r = idx >> 7, c = idx & 127;
      size_t g = (size_t)(base + r) * HD + c;
      As[r * TLD + c] = agg[g];
      Hs[r * TLD + c] = hin[g];
    }
  } else {
    for (int idx = tid; idx < TN * HD; idx += 64) {
      int r = idx >> 7, c = idx & 127;
      int node = base + r;
      float av = 0.f, hv = 0.f;
      if (node < N_NODES) {
        av = agg[(size_t)node * HD + c];
        hv = hin[(size_t)node * HD + c];
      }
      As[r * TLD + c] = av;
      Hs[r * TLD + c] = hv;
    }
  }
  __syncthreads();
  const int lane = tid & 31, wv = tid >> 5;
  const int m = lane & 15, kh = lane >> 4;
  const int wrow = wv * 16;
  v8f acc[8];
#pragma unroll
  for (int j = 0; j < 8; ++j)
#pragma unroll
    for (int i = 0; i < 8; ++i) acc[j][i] = 0.f;
  // y = agg@Wl ; then scale rows by 1/deg (diag row-scaling commutes with GEMM), add bias
  wmma_gemm_f32<8>(As + wrow * TLD, TLD, Wl, HD, acc);
#pragma unroll
  for (int i = 0; i < 8; ++i) {
    float sc = invd[wrow + i + 8 * kh];
#pragma unroll
    for (int j = 0; j < 8; ++j) acc[j][i] = acc[j][i] * sc + bl[j * 16 + m];
  }
  // y += h@Wr
  wmma_gemm_f32<8>(Hs + wrow * TLD, TLD, Wr, HD, acc);

  // Register-resident LayerNorm: acc[j][i] = y[row = wrow+i+8*kh][col = j*16+m].
  // Summing over j then xor-shuffling masks 8..1 reduces over the 16 lanes of each
  // half-wave => full 128-col reduction for that half's row. No LDS round trip.
  float gj[8], bj[8];
#pragma unroll
  for (int j = 0; j < 8; ++j) {
    gj[j] = lng[j * 16 + m];
    bj[j] = lnb[j * 16 + m];
  }
#pragma unroll
  for (int i = 0; i < 8; ++i) {
    float s = 0.f, s2 = 0.f;
#pragma unroll
    for (int j = 0; j < 8; ++j) {
      float v = acc[j][i];
      s += v;
      s2 = fmaf(v, v, s2);
    }
#pragma unroll
    for (int off = 8; off; off >>= 1) {
      s += __shfl_xor(s, off, 32);
      s2 += __shfl_xor(s2, off, 32);
    }
    float mu = s * (1.f / HD);
    float var = s2 * (1.f / HD) - mu * mu;
    float rs = rsqrtf(var + 1e-5f);
    int row = wrow + i + 8 * kh;
    if (full) {
      size_t node = base + row;
#pragma unroll
      for (int j = 0; j < 8; ++j) {
        float yv = (acc[j][i] - mu) * rs * gj[j] + bj[j];
        hout[node * HD + j * 16 + m] = Hs[row * TLD + j * 16 + m] + fmaxf(yv, 0.f);
      }
    } else {
      int node = base + row;
      if (node < N_NODES) {
#pragma unroll
        for (int j = 0; j < 8; ++j) {
          float yv = (acc[j][i] - mu) * rs * gj[j] + bj[j];
          hout[(size_t)node * HD + j * 16 + m] = Hs[row * TLD + j * 16 + m] + fmaxf(yv, 0.f);
        }
      }
    }
  }
}

// ---------------- heads: two 128->128->64->{3,1} MLPs, scaled by ratio ----------------
__global__ __launch_bounds__(64) void heads_kernel(
    const float* __restrict__ h, const float* __restrict__ dW1, const float* __restrict__ db1,
    const float* __restrict__ dW2, const float* __restrict__ db2, const float* __restrict__ dW3,
    const float* __restrict__ db3, const float* __restrict__ sW1, const float* __restrict__ sb1,
    const float* __restrict__ sW2, const float* __restrict__ sb2, const float* __restrict__ sW3,
    const float* __restrict__ sb3, const float* __restrict__ ratio_ws, float* __restrict__ out) {
  __shared__ float Hs[TN * TLD];
  __shared__ float T1[TN * TLD];
  __shared__ float T2[TN * T2LD];
  const int tid = threadIdx.x;
  const int base = blockIdx.x * TN;
  const bool full = (base + TN) <= N_NODES;
  const float ratio = ratio_ws[0];
  if (full) {
    for (int idx = tid; idx < TN * HD; idx += 64) {
      int r = idx >> 7, c = idx & 127;
      Hs[r * TLD + c] = h[(size_t)(base + r) * HD + c];
    }
  } else {
    for (int idx = tid; idx < TN * HD; idx += 64) {
      int r = idx >> 7, c = idx & 127;
      int node = base + r;
      Hs[r * TLD + c] = (node < N_NODES) ? h[(size_t)node * HD + c] : 0.f;
    }
  }
  __syncthreads();
  const int lane = tid & 31, wv = tid >> 5;
  const int m = lane & 15, kh = lane >> 4;
  const int wrow = wv * 16;

  for (int head = 0; head < 2; ++head) {
    const float* W1p = head ? sW1 : dW1;
    const float* b1p = head ? sb1 : db1;
    const float* W2p = head ? sW2 : dW2;
    const float* b2p = head ? sb2 : db2;
    const float* W3p = head ? sW3 : dW3;
    const float* b3p = head ? sb3 : db3;

    v8f a1[8];
#pragma unroll
    for (int j = 0; j < 8; ++j) {
      float bv = b1p[j * 16 + m];
#pragma unroll
      for (int i = 0; i < 8; ++i) a1[j][i] = bv;
    }
    wmma_gemm_f32<8>(Hs + wrow * TLD, TLD, W1p, HD, a1);
#pragma unroll
    for (int j = 0; j < 8; ++j)
#pragma unroll
      for (int i = 0; i < 8; ++i)
        T1[(wrow + i + 8 * kh) * TLD + j * 16 + m] = fmaxf(a1[j][i], 0.f);
    __syncthreads();

    v8f a2[4];
#pragma unroll
    for (int j = 0; j < 4; ++j) {
      float bv = b2p[j * 16 + m];
#pragma unroll
      for (int i = 0; i < 8; ++i) a2[j][i] = bv;
    }
    wmma_gemm_f32<4>(T1 + wrow * TLD, TLD, W2p, 64, a2);
#pragma unroll
    for (int j = 0; j < 4; ++j)
#pragma unroll
      for (int i = 0; i < 8; ++i)
        T2[(wrow + i + 8 * kh) * T2LD + j * 16 + m] = fmaxf(a2[j][i], 0.f);
    __syncthreads();

    // tiny final matmul: 64 -> (3 or 1); lane pair splits K range
    const int r = lane & 15, half = lane >> 4;
    const float* t2r = T2 + (wrow + r) * T2LD + half * 32;
    const int node = base + wrow + r;
    if (head == 0) {
      float p0 = 0.f, p1 = 0.f, p2 = 0.f;
#pragma unroll 4
      for (int k = 0; k < 32; ++k) {
        float tv = t2r[k];
        int kk = half * 32 + k;
        p0 = fmaf(tv, W3p[kk * 3 + 0], p0);
        p1 = fmaf(tv, W3p[kk * 3 + 1], p1);
        p2 = fmaf(tv, W3p[kk * 3 + 2], p2);
      }
      p0 += __shfl_xor(p0, 16, 32);
      p1 += __shfl_xor(p1, 16, 32);
      p2 += __shfl_xor(p2, 16, 32);
      if (half == 0 && node < N_NODES) {
        out[(size_t)node * 4 + 0] = (p0 + b3p[0]) * ratio;
        out[(size_t)node * 4 + 1] = (p1 + b3p[1]) * ratio;
        out[(size_t)node * 4 + 2] = (p2 + b3p[2]) * ratio;
      }
    } else {
      float p0 = 0.f;
#pragma unroll 4
      for (int k = 0; k < 32; ++k) p0 = fmaf(t2r[k], W3p[half * 32 + k], p0);
      p0 += __shfl_xor(p0, 16, 32);
      if (half == 0 && node < N_NODES) out[(size_t)node * 4 + 3] = (p0 + b3p[0]) * ratio;
    }
    __syncthreads();  // T1/T2 reused by next head
  }
}

extern "C" void kernel_launch(void* const* d_in, const int* in_sizes, int n_in,
                              void* d_out, int out_size, void* d_ws, size_t ws_size,
                              hipStream_t stream) {
  const float* x = (const float*)d_in[0];
  const int* ei = (const int*)d_in[1];
  const float* enc_W1 = (const float*)d_in[2];
  const float* enc_b1 = (const float*)d_in[3];
  const float* enc_W2 = (const float*)d_in[4];
  const float* enc_b2 = (const float*)d_in[5];
  const float* lin_l_W = (const float*)d_in[6];
  const float* lin_l_b = (const float*)d_in[7];
  const float* lin_r_W = (const float*)d_in[8];
  const float* ln_g = (const float*)d_in[9];
  const float* ln_b = (const float*)d_in[10];
  const float* dW1 = (const float*)d_in[11];
  const float* db1 = (const float*)d_in[12];
  const float* dW2 = (const float*)d_in[13];
  const float* db2 = (const float*)d_in[14];
  const float* dW3 = (const float*)d_in[15];
  const float* db3 = (const float*)d_in[16];
  const float* sW1 = (const float*)d_in[17];
  const float* sb1 = (const float*)d_in[18];
  const float* sW2 = (const float*)d_in[19];
  const float* sb2 = (const float*)d_in[20];
  const float* sW3 = (const float*)d_in[21];
  const float* sb3 = (const float*)d_in[22];
  float* out = (float*)d_out;

  float* ws = (float*)d_ws;
  const size_t NH = (size_t)N_NODES * HD;
  float* ratio = ws;                 // 64 floats
  float* deg = ws + 64;              // 50048 floats
  float* agg = ws + 64 + 50048;      // NH
  float* h_a = agg + NH;             // NH
  float* h_b = h_a + NH;             // NH

  hipMemsetAsync(d_ws, 0, (64 + 50048) * sizeof(float), stream);

  ratio_kernel<<<(N_NODES + 255) / 256, 256, 0, stream>>>(x, ratio);
  degree_kernel<<<(N_EDGES + 255) / 256, 256, 0, stream>>>(ei, deg);

  const int nblk = (N_NODES + TN - 1) / TN;
  encoder_kernel<<<nblk, 64, 0, stream>>>(x, enc_W1, enc_b1, enc_W2, enc_b2, h_a);

  float* hc = h_a;
  float* hn = h_b;
  for (int l = 0; l < NL; ++l) {
    hipMemsetAsync(agg, 0, NH * sizeof(float), stream);
    scatter_kernel<<<(N_EDGES * 32 + 255) / 256, 256, 0, stream>>>(hc, ei, agg);
    sage_kernel<<<nblk, 64, 0, stream>>>(hc, agg, deg, lin_l_W + (size_t)l * HD * HD,
                                         lin_l_b + l * HD, lin_r_W + (size_t)l * HD * HD,
                                         ln_g + l * HD, ln_b + l * HD, hn);
    float* t = hc;
    hc = hn;
    hn = t;
  }

  heads_kernel<<<nblk, 64, 0, stream>>>(hc, dW1, db1, dW2, db2, dW3, db3, sW1, sb1, sW2, sb2,
                                        sW3, sb3, ratio, out);
}